// EquivariantBinaryClassificationSAGPoolScalar_39513699123761
// MI455X (gfx1250) — compile-verified
//
#include <hip/hip_runtime.h>
#include <hip/hip_bf16.h>

// ---- problem constants (match reference) -----------------------------------
#define NB    64        // graphs
#define NPG1  1024      // nodes per graph, layer-0 input
#define CCH   256       // node channels
#define ECH   48        // edge channels
#define NE    1048576   // edges  (NB*NPG1*16)
#define NN    65536     // nodes  (NB*NPG1)
#define KP1   512       // kept per graph after pool 1
#define NK1   32768     // NB*KP1
#define KP2   256       // kept per graph after pool 2
#define NK2   16384     // NB*KP2

typedef float v2f __attribute__((ext_vector_type(2)));
typedef float v8f __attribute__((ext_vector_type(8)));

// ---------------------------------------------------------------------------
// Mat-vec via V_WMMA_F32_16X16X4_F32.
// One wave computes 16 rows of  X[rows,K] @ w0 (+b0)  and (if DUAL) @ w1 (+b1)
// by placing w0 in column 0 and w1 in column 1 of the 4x16 B operand.
// B construction is branchless (selector floats) so the inner loop carries no
// EXEC manipulation: load A (b64) -> build B (2 FMA) -> v_wmma.
// rows % 16 == 0 and K % 4 == 0 at every call site.
// ---------------------------------------------------------------------------
template <bool DUAL>
__global__ void matvec_wmma(const float* __restrict__ X,
                            const float* __restrict__ w0, const float* __restrict__ b0,
                            float* __restrict__ out0,
                            const float* __restrict__ w1, const float* __restrict__ b1,
                            float* __restrict__ out1,
                            int rows, int K) {
  const int lane = threadIdx.x & 31;
  const int wave = threadIdx.x >> 5;
  const long tile = (long)blockIdx.x * (blockDim.x >> 5) + wave;
  const long row0 = tile * 16;
  if (row0 >= rows) return;                 // wave-uniform: EXEC stays all-ones
  const int half = lane >> 4;               // 0 or 1
  const int m    = lane & 15;

  // selector floats -> pure VALU B-fragment build, no branches in the loop
  const float sel0 = (m == 0) ? 1.f : 0.f;
  const float sel1 = (DUAL && m == 1) ? 1.f : 0.f;
  const float* __restrict__ wB = DUAL ? w1 : w0;  // always a valid pointer

  v8f c = (v8f){0.f,0.f,0.f,0.f,0.f,0.f,0.f,0.f};
  const float* xr = X + (row0 + m) * (long)K + 2 * half;
  for (int k0 = 0; k0 < K; k0 += 4) {
    v2f a, b;
    a.x = xr[k0];                           // A[m, k0+2h]
    a.y = xr[k0 + 1];                       // A[m, k0+2h+1]
    // B[k][n]: column 0 = w0, column 1 = w1 (if DUAL), rest 0
    b.x = sel0 * w0[k0 + half]     + sel1 * wB[k0 + half];
    b.y = sel0 * w0[k0 + 2 + half] + sel1 * wB[k0 + 2 + half];
    c = __builtin_amdgcn_wmma_f32_16x16x4_f32(false, a, false, b,
                                              (short)0, c, false, false);
  }
  // D column 0 -> lanes 0 (M=0..7 in c[0..7]) and 16 (M=8..15); column 1 -> lanes 1/17
  if (m == 0) {
    float bv = b0[0];
    #pragma unroll
    for (int r = 0; r < 8; ++r) out0[row0 + half * 8 + r] = c[r] + bv;
  } else if (DUAL && m == 1) {
    float bv = b1[0];
    #pragma unroll
    for (int r = 0; r < 8; ++r) out1[row0 + half * 8 + r] = c[r] + bv;
  }
}

// ---------------------------------------------------------------------------
// Edge scatter:  num[dst] += xs[src]*W[e] ; cnt[dst] += 1   (optionally
// remapped through node_map with -1 = dropped, which implements filter_adj).
// ---------------------------------------------------------------------------
__global__ void msg_pass(const int* __restrict__ src, const int* __restrict__ dst,
                         const float* __restrict__ W, const float* __restrict__ xs,
                         const int* __restrict__ node_map,
                         float* __restrict__ num, float* __restrict__ cnt, int nE) {
  int e = blockIdx.x * blockDim.x + threadIdx.x;
  if (e >= nE) return;
  int s = src[e], d = dst[e];
  if (node_map) {
    s = node_map[s]; d = node_map[d];
    if (s < 0 || d < 0) return;
  }
  atomicAdd(&num[d], xs[s] * W[e]);
  atomicAdd(&cnt[d], 1.0f);
}

__global__ void finalize_score(const float* __restrict__ num, const float* __restrict__ cnt,
                               float* __restrict__ score, int n) {
  int i = blockIdx.x * blockDim.x + threadIdx.x;
  if (i < n) score[i] = (cnt[i] > 0.f) ? num[i] / fmaxf(cnt[i], 1.f) : 0.f;
}

// ---------------------------------------------------------------------------
// Per-graph top-k by ranking (exactly matches jax.lax.top_k ordering:
// rank = #{s_j > s_i} + #{s_j == s_i, j < i}).  One block per graph.
// node_map[orig] = g*k + rank (or -1); perm[g*k+rank] = orig.
// ---------------------------------------------------------------------------
__global__ void topk_rank(const float* __restrict__ score, int npg, int k,
                          int* __restrict__ node_map, int* __restrict__ perm) {
  __shared__ float s_sc[NPG1];
  const int g = blockIdx.x;
  const float* sc = score + (long)g * npg;
  for (int i = threadIdx.x; i < npg; i += blockDim.x) s_sc[i] = sc[i];
  __syncthreads();
  for (int i = threadIdx.x; i < npg; i += blockDim.x) {
    float si = s_sc[i];
    int r = 0;
    for (int j = 0; j < npg; ++j) {
      float sj = s_sc[j];
      r += (sj > si) || (sj == si && j < i);
    }
    int node = g * npg + i;
    if (r < k) { node_map[node] = g * k + r; perm[g * k + r] = node; }
    else       { node_map[node] = -1; }
  }
}

// x_new[j] = x[perm[j]] * tanh(score[perm[j]])  ; one block per kept node
__global__ void gather_gate(const float* __restrict__ X, const float* __restrict__ score,
                            const int* __restrict__ perm, float* __restrict__ Y, int C_) {
  const int j = blockIdx.x;
  const int o = perm[j];
  const float gate = tanhf(score[o]);
  const float* xr = X + (long)o * C_;
  float* yr = Y + (long)j * C_;
  for (int c = threadIdx.x; c < C_; c += blockDim.x) yr[c] = xr[c] * gate;
}

// outacc[g][c] (=|+=) sum_r X[g*rpg+r][c]
__global__ void segsum(const float* __restrict__ X, float* __restrict__ outacc,
                       int rpg, int C_, int do_add) {
  const int g = blockIdx.x, c = threadIdx.x;
  float acc = 0.f;
  const float* p = X + ((long)g * rpg) * C_ + c;
  for (int r = 0; r < rpg; ++r) acc += p[(long)r * C_];
  float* o = &outacc[g * C_ + c];
  *o = do_add ? (*o + acc) : acc;
}

// outacc[g][c] += sum_r x1[perm2[g*kpg+r]][c] * tanh(score2[perm2[...]])
__global__ void segsum_gate_add(const float* __restrict__ X, const int* __restrict__ perm2,
                                const float* __restrict__ score2, float* __restrict__ outacc,
                                int kpg, int C_) {
  const int g = blockIdx.x, c = threadIdx.x;
  float acc = 0.f;
  for (int r = 0; r < kpg; ++r) {
    int o = perm2[g * kpg + r];
    acc += X[(long)o * C_ + c] * tanhf(score2[o]);
  }
  outacc[g * C_ + c] += acc;
}

__global__ void sigmoid_k(const float* __restrict__ t, float* __restrict__ out, int n) {
  int i = blockIdx.x * blockDim.x + threadIdx.x;
  if (i < n) out[i] = 1.f / (1.f + expf(-t[i]));
}

// ---------------------------------------------------------------------------
extern "C" void kernel_launch(void* const* d_in, const int* in_sizes, int n_in,
                              void* d_out, int out_size, void* d_ws, size_t ws_size,
                              hipStream_t stream) {
  const float* x     = (const float*)d_in[0];
  const int*   ei    = (const int*)  d_in[1];
  const float* ea    = (const float*)d_in[2];
  // d_in[3] = batch (implicit: node i -> graph i/NPG1)
  const float* dp_w1 = (const float*)d_in[4];
  const float* dp_b1 = (const float*)d_in[5];
  const float* sc_w1 = (const float*)d_in[6];
  const float* sc_b1 = (const float*)d_in[7];
  const float* dp_w2 = (const float*)d_in[8];
  const float* dp_b2 = (const float*)d_in[9];
  const float* sc_w2 = (const float*)d_in[10];
  const float* sc_b2 = (const float*)d_in[11];
  const float* out_w = (const float*)d_in[12];
  const float* out_b = (const float*)d_in[13];
  float* out = (float*)d_out;

  const int* src = ei;          // edge_index[0]
  const int* dst = ei + NE;     // edge_index[1]

  // ---- workspace carve-up (256B aligned; everything read is rewritten) ----
  char* p = (char*)d_ws;
  auto carve = [&](size_t bytes) { char* q = p; p += (bytes + 255) & ~(size_t)255; return q; };
  float* W1      = (float*)carve((size_t)NE  * 4);
  float* W2      = (float*)carve((size_t)NE  * 4);
  float* xs1     = (float*)carve((size_t)NN  * 4);
  float* num1    = (float*)carve((size_t)NN  * 4);   // contiguous with cnt1
  float* cnt1    = (float*)carve((size_t)NN  * 4);
  float* score1  = (float*)carve((size_t)NN  * 4);
  int*   nmap1   = (int*)  carve((size_t)NN  * 4);
  int*   perm1   = (int*)  carve((size_t)NK1 * 4);
  float* x1      = (float*)carve((size_t)NK1 * CCH * 4);
  float* xs2     = (float*)carve((size_t)NK1 * 4);
  float* num2    = (float*)carve((size_t)NK1 * 4);   // contiguous with cnt2
  float* cnt2    = (float*)carve((size_t)NK1 * 4);
  float* score2  = (float*)carve((size_t)NK1 * 4);
  int*   nmap2   = (int*)  carve((size_t)NK1 * 4);   // unused downstream
  int*   perm2   = (int*)  carve((size_t)NK2 * 4);
  float* outacc  = (float*)carve((size_t)NB * CCH * 4);
  float* tmp     = (float*)carve((size_t)64 * 4);

  hipMemsetAsync(num1, 0, 2 * (size_t)NN  * 4, stream);  // num1 + cnt1
  hipMemsetAsync(num2, 0, 2 * (size_t)NK1 * 4, stream);  // num2 + cnt2

  // Both edge projections in ONE pass over edge_attr (192MB read once).
  matvec_wmma<true><<<NE / 128, 256, 0, stream>>>(ea, dp_w1, dp_b1, W1,
                                                  dp_w2, dp_b2, W2, NE, ECH);
  // xs1 = x @ sc_w1 + b
  matvec_wmma<false><<<NN / 128, 256, 0, stream>>>(x, sc_w1, sc_b1, xs1,
                                                   nullptr, nullptr, nullptr, NN, CCH);
  // out = segment_sum(x)
  segsum<<<NB, CCH, 0, stream>>>(x, outacc, NPG1, CCH, 0);

  // ---- layer 1 ----
  msg_pass<<<NE / 256, 256, 0, stream>>>(src, dst, W1, xs1, nullptr, num1, cnt1, NE);
  finalize_score<<<NN / 256, 256, 0, stream>>>(num1, cnt1, score1, NN);
  topk_rank<<<NB, 256, 0, stream>>>(score1, NPG1, KP1, nmap1, perm1);
  gather_gate<<<NK1, 256, 0, stream>>>(x, score1, perm1, x1, CCH);
  segsum<<<NB, CCH, 0, stream>>>(x1, outacc, KP1, CCH, 1);

  // ---- layer 2 (edges filtered on the fly via nmap1) ----
  matvec_wmma<false><<<NK1 / 128, 256, 0, stream>>>(x1, sc_w2, sc_b2, xs2,
                                                    nullptr, nullptr, nullptr, NK1, CCH);
  msg_pass<<<NE / 256, 256, 0, stream>>>(src, dst, W2, xs2, nmap1, num2, cnt2, NE);
  finalize_score<<<NK1 / 256, 256, 0, stream>>>(num2, cnt2, score2, NK1);
  topk_rank<<<NB, 256, 0, stream>>>(score2, KP1, KP2, nmap2, perm2);
  segsum_gate_add<<<NB, CCH, 0, stream>>>(x1, perm2, score2, outacc, KP2, CCH);

  // ---- head: sigmoid(outacc @ out_w + out_b) ----
  matvec_wmma<false><<<1, 256, 0, stream>>>(outacc, out_w, out_b, tmp,
                                            nullptr, nullptr, nullptr, NB, CCH);
  sigmoid_k<<<1, 64, 0, stream>>>(tmp, out, NB);
}